// CodeLayer_28449863369503
// MI455X (gfx1250) — compile-verified
//
#include <hip/hip_runtime.h>

// ---------------------------------------------------------------------------
// VQ-VAE quantize + EMA stats for MI455X (gfx1250, wave32, WMMA)
//
// Score GEMM uses V_WMMA_F32_16X16X4_F32: A = 16 tokens x 4 dims (f32),
// B = 4 dims x 16 codes (f32, from LDS-staged transposed codebook),
// C = 16x16 f32 scores. argmin(dist) == argmax(x.E - 0.5*||E||^2).
// ---------------------------------------------------------------------------

typedef float v2f __attribute__((ext_vector_type(2)));
typedef float v8f __attribute__((ext_vector_type(8)));

#define N_TOK    131072
#define DIM      64
#define N_EMB    1024
#define ROW      68            // padded row stride (floats) for LDS codebook: bank-conflict free b64 reads
#define DECAYF   0.99f
#define ONEMD    0.01f
#define EPSF     1e-5f

#define MAIN_WGS     256
#define WAVES_PER_WG 8

#define SMEM_EMBED_FLOATS (N_EMB * ROW)                    // 69632
#define SMEM_NORM_OFF     SMEM_EMBED_FLOATS                // 1024 floats (0.5*||E||^2)
#define SMEM_WIDX_OFF     (SMEM_NORM_OFF + N_EMB)          // 8 waves * 16 ints
#define SMEM_RED_OFF      (SMEM_WIDX_OFF + WAVES_PER_WG*16)
#define SMEM_FLOATS       (SMEM_RED_OFF + WAVES_PER_WG)
#define SMEM_BYTES        (SMEM_FLOATS * 4)                // ~283 KB < 320 KB WGP LDS

// ---------------------------------------------------------------------------
__global__ void vq_zero_ws(float* ws, int n) {
  int i = blockIdx.x * blockDim.x + threadIdx.x;
  if (i < n) ws[i] = 0.0f;
}

// ---------------------------------------------------------------------------
__global__ __launch_bounds__(256) void vq_main(
    const float* __restrict__ x,       // [N_TOK, DIM]
    const float* __restrict__ embed,   // [DIM, N_EMB]
    float* __restrict__ out_q,         // [N_TOK, DIM] quantize_st
    float* __restrict__ out_ind,       // [N_TOK] argmin index (stored as float)
    float* __restrict__ ws_cnt,        // [N_EMB] onehot_sum accum
    float* __restrict__ ws_esum,       // [DIM, N_EMB] embed_sum accum
    float* __restrict__ ws_diff)       // [1] sum of (q-x)^2
{
  extern __shared__ float smem[];
  float* et   = smem;                        // codebook, transposed: et[e*ROW + d]
  float* nrm  = smem + SMEM_NORM_OFF;        // 0.5*||E_e||^2
  int*   widx = (int*)(smem + SMEM_WIDX_OFF);
  float* red  = smem + SMEM_RED_OFF;

  const int tid = threadIdx.x;

  // Stage codebook into LDS, transposed + padded (global reads coalesced).
  for (int i = tid; i < DIM * N_EMB; i += blockDim.x) {
    int d = i >> 10;          // DIM index
    int e = i & (N_EMB - 1);  // code index
    et[e * ROW + d] = embed[i];
  }
  __syncthreads();
  // Precompute 0.5*||E||^2 per code.
  for (int e = tid; e < N_EMB; e += blockDim.x) {
    float s = 0.0f;
    #pragma unroll
    for (int d = 0; d < DIM; ++d) { float v = et[e * ROW + d]; s += v * v; }
    nrm[e] = 0.5f * s;
  }
  __syncthreads();

  const int lane = tid & 31;
  const int wave = tid >> 5;
  const int half = lane >> 4;   // 0: lanes 0-15, 1: lanes 16-31
  const int m    = lane & 15;   // token-within-tile (A rows) / code-within-tile (B cols)

  float diff_acc = 0.0f;

  const int n_tiles = N_TOK / 16;
  const int wstride = gridDim.x * WAVES_PER_WG;
  for (int tile = blockIdx.x * WAVES_PER_WG + wave; tile < n_tiles; tile += wstride) {
    const int t0 = tile * 16;

    // Load all A fragments for this 16-token tile (kept in VGPRs for all 64 code tiles).
    // A 16x4 f32 layout: lane<16 holds (M=lane, K=k0..k0+1), lane>=16 holds (M=lane-16, K=k0+2..k0+3).
    v2f a[16];
    #pragma unroll
    for (int k = 0; k < 16; ++k) {
      const int d0 = 4 * k + 2 * half;
      a[k] = *(const v2f*)(x + (size_t)(t0 + m) * DIM + d0);
    }

    float bestv[8];
    int   besti[8];
    #pragma unroll
    for (int v = 0; v < 8; ++v) { bestv[v] = -3.4e38f; besti[v] = 0; }

    for (int n = 0; n < N_EMB / 16; ++n) {
      const int N0 = n * 16 + m;                       // this lane's code column
      const float* bp = et + N0 * ROW + 2 * half;      // B fragment base (mirrors A layout)
      v8f c = {0.f, 0.f, 0.f, 0.f, 0.f, 0.f, 0.f, 0.f};
      #pragma unroll
      for (int k = 0; k < 16; ++k) {
        v2f b = *(const v2f*)(bp + 4 * k);             // one ds_load_b64, conflict-free (ROW=68)
        c = __builtin_amdgcn_wmma_f32_16x16x4_f32(
                false, a[k], false, b, (short)0, c, false, false);
      }
      const float hn = nrm[N0];
      #pragma unroll
      for (int v = 0; v < 8; ++v) {
        float s = c[v] - hn;                           // = x.E - 0.5*||E||^2 ; maximize
        if (s > bestv[v]) { bestv[v] = s; besti[v] = N0; }  // per-lane indices increase -> first-wins
      }
    }

    // Reduce across the 16 lanes holding each C row (argmax, tie -> smaller index == jnp.argmin).
    #pragma unroll
    for (int off = 1; off < 16; off <<= 1) {
      #pragma unroll
      for (int v = 0; v < 8; ++v) {
        float ov = __shfl_xor(bestv[v], off, 16);
        int   oi = __shfl_xor(besti[v], off, 16);
        if (ov > bestv[v] || (ov == bestv[v] && oi < besti[v])) {
          bestv[v] = ov; besti[v] = oi;
        }
      }
    }

    // Rows M=v live in lanes 0-15 (half 0) and M=v+8 in lanes 16-31 (half 1).
    if (m == 0) {
      #pragma unroll
      for (int v = 0; v < 8; ++v) {
        const int tokm = v + 8 * half;
        widx[wave * 16 + tokm] = besti[v];
        out_ind[t0 + tokm] = (float)besti[v];   // harness stores non-bf16 outputs as float
      }
    }
    __builtin_amdgcn_wave_barrier();            // LDS ops are in-order within a wave

    // Gather winning code from LDS, emit straight-through quantize, accumulate diff + EMA scatter.
    const int idxm = widx[wave * 16 + m];       // code index for token (t0+m)
    #pragma unroll
    for (int k = 0; k < 16; ++k) {
      const int d0 = 4 * k + 2 * half;
      v2f q  = *(const v2f*)(et + idxm * ROW + d0);
      v2f xv = a[k];
      float dx = q.x - xv.x;
      float dy = q.y - xv.y;
      diff_acc += dx * dx + dy * dy;
      v2f qst = { xv.x + dx, xv.y + dy };       // x + (quantize - x), matches reference rounding
      *(v2f*)(out_q + (size_t)(t0 + m) * DIM + d0) = qst;
      atomicAdd(ws_esum + (size_t)d0 * N_EMB + idxm,       xv.x);
      atomicAdd(ws_esum + (size_t)(d0 + 1) * N_EMB + idxm, xv.y);
    }
    if (lane < 16) atomicAdd(ws_cnt + idxm, 1.0f);  // one count per token
  }

  // diff reduction: wave -> workgroup -> one global atomic per workgroup
  #pragma unroll
  for (int off = 16; off > 0; off >>= 1) diff_acc += __shfl_xor(diff_acc, off, 32);
  __syncthreads();
  if (lane == 0) red[wave] = diff_acc;
  __syncthreads();
  if (tid == 0) {
    float s = 0.0f;
    #pragma unroll
    for (int w = 0; w < WAVES_PER_WG; ++w) s += red[w];
    atomicAdd(ws_diff, s);
  }
}

// ---------------------------------------------------------------------------
__global__ __launch_bounds__(1024) void vq_finalize(
    const float* __restrict__ cluster_size,  // [N_EMB]
    const float* __restrict__ embed_avg,     // [DIM, N_EMB]
    const float* __restrict__ ws_cnt,
    const float* __restrict__ ws_esum,
    const float* __restrict__ ws_diff,
    float* __restrict__ out_diff,            // [1]
    float* __restrict__ out_new_embed,       // [DIM, N_EMB]
    float* __restrict__ out_ncs,             // [N_EMB]
    float* __restrict__ out_nea)             // [DIM, N_EMB]
{
  __shared__ float sred[1024];
  const int t = threadIdx.x;

  float ncs = cluster_size[t] * DECAYF + ONEMD * ws_cnt[t];
  out_ncs[t] = ncs;
  sred[t] = ncs;
  __syncthreads();
  #pragma unroll
  for (int off = 512; off > 0; off >>= 1) {
    if (t < off) sred[t] += sred[t + off];
    __syncthreads();
  }
  const float n  = sred[0];
  const float cs = (ncs + EPSF) / (n + (float)N_EMB * EPSF) * n;

  for (int d = 0; d < DIM; ++d) {
    const int i = d * N_EMB + t;   // 32 consecutive lanes -> coalesced
    float nea = embed_avg[i] * DECAYF + ONEMD * ws_esum[i];
    out_nea[i] = nea;
    out_new_embed[i] = nea / cs;
  }
  if (t == 0) out_diff[0] = ws_diff[0] * (1.0f / (float)((size_t)N_TOK * DIM));
}

// ---------------------------------------------------------------------------
extern "C" void kernel_launch(void* const* d_in, const int* in_sizes, int n_in,
                              void* d_out, int out_size, void* d_ws, size_t ws_size,
                              hipStream_t stream) {
  (void)in_sizes; (void)n_in; (void)out_size; (void)ws_size;

  const float* x            = (const float*)d_in[0];
  const float* embed        = (const float*)d_in[1];
  const float* cluster_size = (const float*)d_in[2];
  const float* embed_avg    = (const float*)d_in[3];

  // Output tuple, concatenated flat (all float):
  float* out      = (float*)d_out;
  float* out_q    = out;                                   // [N_TOK*DIM]
  float* out_diff = out + (size_t)N_TOK * DIM;             // [1]
  float* out_ind  = out_diff + 1;                          // [N_TOK]
  float* out_ne   = out_ind + N_TOK;                       // [DIM*N_EMB]
  float* out_ncs  = out_ne + (size_t)DIM * N_EMB;          // [N_EMB]
  float* out_nea  = out_ncs + N_EMB;                       // [DIM*N_EMB]

  // Workspace: cnt[1024] | esum[64*1024] | diff[1]
  float* ws      = (float*)d_ws;
  float* ws_cnt  = ws;
  float* ws_esum = ws + N_EMB;
  float* ws_diff = ws + N_EMB + (size_t)DIM * N_EMB;
  const int ws_n = N_EMB + DIM * N_EMB + 1;

  vq_zero_ws<<<(ws_n + 255) / 256, 256, 0, stream>>>(ws, ws_n);

  (void)hipFuncSetAttribute(reinterpret_cast<const void*>(vq_main),
                            hipFuncAttributeMaxDynamicSharedMemorySize, SMEM_BYTES);
  vq_main<<<MAIN_WGS, 256, SMEM_BYTES, stream>>>(
      x, embed, out_q, out_ind, ws_cnt, ws_esum, ws_diff);

  vq_finalize<<<1, 1024, 0, stream>>>(
      cluster_size, embed_avg, ws_cnt, ws_esum, ws_diff,
      out_diff, out_ne, out_ncs, out_nea);
}